// MMPPLoss_56298431316105
// MI455X (gfx1250) — compile-verified
//
#include <hip/hip_runtime.h>
#include <stdint.h>

typedef __attribute__((ext_vector_type(2))) float v2f;
typedef __attribute__((ext_vector_type(4))) float v4f;
typedef __attribute__((ext_vector_type(8))) float v8f;
typedef __attribute__((ext_vector_type(4))) int   v4i;

// ---------------------------------------------------------------------------
// Wave32 sum via V_WMMA_F32_16X16X4_F32.
// A layout (ISA 7.12.2, 32-bit A 16x4): lane L<16 holds A[L][K=0] in VGPR0,
// A[L][K=1] in VGPR1; lane L>=16 holds A[L-16][K=2], A[L-16][K=3].
// With A = {acc, 0} and B = all-ones, D[m][n] = acc_m + acc_{m+16}.
// D layout: lane n (0-15) holds D[0..7][n], lane n+16 holds D[8..15][n].
// Sum of the 8 D components per lane = half the wave sum; xor-16 shuffle
// combines the halves. EXEC must be all ones (fully-converged call site).
// ---------------------------------------------------------------------------
__device__ __forceinline__ float wave_sum_wmma(float acc) {
  v2f a; a.x = acc;  a.y = 0.0f;
  v2f bm; bm.x = 1.0f; bm.y = 1.0f;
  v8f c = {};
  v8f d = __builtin_amdgcn_wmma_f32_16x16x4_f32(
      /*neg_a=*/false, a, /*neg_b=*/false, bm,
      /*c_mod=*/(short)0, c, /*reuse_a=*/false, /*reuse_b=*/false);
  float s = d[0] + d[1] + d[2] + d[3] + d[4] + d[5] + d[6] + d[7];
  s += __shfl_xor(s, 16, 32);   // combine the two 16-row halves
  return s;                      // full wave sum in every lane
}

// Block of exactly 256 threads (8 wave32s).
__device__ __forceinline__ float block_sum256(float acc, float* sp) {
  float w = wave_sum_wmma(acc);
  int t = threadIdx.x;
  if ((t & 31) == 0) sp[t >> 5] = w;
  __syncthreads();
  float s = 0.0f;
  if (t == 0) {
#pragma unroll
    for (int i = 0; i < 8; ++i) s += sp[i];
  }
  return s;  // valid on thread 0 only
}

__device__ __forceinline__ float bin_term(int m, int bb, int st, int ds, float du,
                                          const float* aH, const float* aA,
                                          const float* sb, const float* sgH,
                                          const float* sgA, const float* sdH,
                                          const float* sdA) {
  float base = sb[bb];
  float lh = aH[m] + base + sgH[st] + sdH[ds];
  float la = aA[m] + base + sgA[st] + sdA[ds];
  return (__expf(lh) + __expf(la)) * du;
}

__device__ __forceinline__ float goal_term(int m, int bb, int st, int ds, bool home,
                                           const float* aH, const float* aA,
                                           const float* sb, const float* sgH,
                                           const float* sgA, const float* sdH,
                                           const float* sdA) {
  const float* aa = home ? aH : aA;
  float g  = home ? sgH[st] : sgA[st];
  float dd = home ? sdH[ds] : sdA[ds];
  return aa[m] + sb[bb] + g + dd;
}

// ---------------------------------------------------------------------------
// Kernel 1: grid-stride streamer over bins + goals + reg. All single-use
// streams are loaded non-temporally (b128, TH=NT) so the L2 stays warm for
// the random a_H/a_A gathers (2 MB each, resident in the 192 MB L2).
// One f32 partial per block goes to d_ws. ~242 MB total -> ~10 us roofline
// at 23.3 TB/s; compute (20M v_exp_f32) is negligible next to that.
// ---------------------------------------------------------------------------
__global__ __launch_bounds__(256) void mmpp_partial(
    const float* __restrict__ a_H, const float* __restrict__ a_A,
    const float* __restrict__ a_H_init, const float* __restrict__ a_A_init,
    const float* __restrict__ b, const float* __restrict__ gamma_H,
    const float* __restrict__ gamma_A, const float* __restrict__ delta_H,
    const float* __restrict__ delta_A,
    const int* __restrict__ bmi, const int* __restrict__ bbi,
    const int* __restrict__ bsi, const int* __restrict__ bdi,
    const float* __restrict__ bdur,
    const int* __restrict__ gmi, const int* __restrict__ gbi,
    const int* __restrict__ gsi, const int* __restrict__ gdi,
    const unsigned char* __restrict__ gih,
    int NB, int NG, int M, float* __restrict__ block_part) {
  __shared__ float s_b[6], s_gH[4], s_gA[4], s_dH[5], s_dA[5];
  __shared__ float s_part[8];
  int t = threadIdx.x;
  if (t < 6) s_b[t] = b[t];
  if (t < 4) {
    s_gH[t] = gamma_H[0] * (float)(t & 1) + gamma_H[1] * (float)(t >> 1);
    s_gA[t] = gamma_A[0] * (float)(t & 1) + gamma_A[1] * (float)(t >> 1);
  }
  if (t < 5) {
    s_dH[t] = t ? delta_H[t - 1] : 0.0f;
    s_dA[t] = t ? delta_A[t - 1] : 0.0f;
  }
  __syncthreads();

  const int gsize = gridDim.x * blockDim.x;
  const int gtid  = blockIdx.x * blockDim.x + t;
  float acc = 0.0f;

  // ---- exposure terms: + sum((exp(lh)+exp(la))*dur), vec4 NT streams ----
  int nb4 = NB >> 2;
  for (int i = gtid; i < nb4; i += gsize) {
    v4i m  = __builtin_nontemporal_load((const v4i*)bmi + i);
    v4i bb = __builtin_nontemporal_load((const v4i*)bbi + i);
    v4i st = __builtin_nontemporal_load((const v4i*)bsi + i);
    v4i ds = __builtin_nontemporal_load((const v4i*)bdi + i);
    v4f du = __builtin_nontemporal_load((const v4f*)bdur + i);
#pragma unroll
    for (int k = 0; k < 4; ++k)
      acc += bin_term(m[k], bb[k], st[k], ds[k], du[k],
                      a_H, a_A, s_b, s_gH, s_gA, s_dH, s_dA);
  }
  for (int i = (nb4 << 2) + gtid; i < NB; i += gsize)
    acc += bin_term(bmi[i], bbi[i], bsi[i], bdi[i], bdur[i],
                    a_H, a_A, s_b, s_gH, s_gA, s_dH, s_dA);

  // ---- goal terms: - sum(ln_lambda), vec4 (bools packed 4-per-uint) ----
  int ng4 = NG >> 2;
  for (int i = gtid; i < ng4; i += gsize) {
    v4i m  = __builtin_nontemporal_load((const v4i*)gmi + i);
    v4i bb = __builtin_nontemporal_load((const v4i*)gbi + i);
    v4i st = __builtin_nontemporal_load((const v4i*)gsi + i);
    v4i ds = __builtin_nontemporal_load((const v4i*)gdi + i);
    uint32_t hm = __builtin_nontemporal_load((const uint32_t*)gih + i);
#pragma unroll
    for (int k = 0; k < 4; ++k)
      acc -= goal_term(m[k], bb[k], st[k], ds[k],
                       ((hm >> (8 * k)) & 0xffu) != 0,
                       a_H, a_A, s_b, s_gH, s_gA, s_dH, s_dA);
  }
  for (int i = (ng4 << 2) + gtid; i < NG; i += gsize)
    acc -= goal_term(gmi[i], gbi[i], gsi[i], gdi[i], gih[i] != 0,
                     a_H, a_A, s_b, s_gH, s_gA, s_dH, s_dA);

  // ---- gaussian prior on a: + 0.5*sum((a-a0)^2), vec4 ----
  // (a_H/a_A loaded temporally here too -- they are the L2-resident tables.)
  int m4 = M >> 2;
  for (int i = gtid; i < m4; i += gsize) {
    v4f h  = *((const v4f*)a_H + i);
    v4f h0 = __builtin_nontemporal_load((const v4f*)a_H_init + i);
    v4f aa = *((const v4f*)a_A + i);
    v4f a0 = __builtin_nontemporal_load((const v4f*)a_A_init + i);
    float r = 0.0f;
#pragma unroll
    for (int k = 0; k < 4; ++k) {
      float dh = h[k] - h0[k];
      float da = aa[k] - a0[k];
      r += dh * dh + da * da;
    }
    acc += 0.5f * r;
  }
  for (int i = (m4 << 2) + gtid; i < M; i += gsize) {
    float dh = a_H[i] - a_H_init[i];
    float da = a_A[i] - a_A_init[i];
    acc += 0.5f * (dh * dh + da * da);
  }

  float s = block_sum256(acc, s_part);   // WMMA wave reduce inside
  if (t == 0) block_part[blockIdx.x] = s;
}

// ---------------------------------------------------------------------------
// Kernel 2: one block reduces the block partials + adds the tiny L2 term.
// ---------------------------------------------------------------------------
__global__ __launch_bounds__(256) void mmpp_final(
    const float* __restrict__ part, int nparts,
    const float* __restrict__ b, const float* __restrict__ gH,
    const float* __restrict__ gA, const float* __restrict__ dH,
    const float* __restrict__ dA, float* __restrict__ out) {
  __shared__ float s_part[8];
  float acc = 0.0f;
  for (int i = threadIdx.x; i < nparts; i += blockDim.x) acc += part[i];
  if (threadIdx.x == 0) {
    float r = 0.0f;
#pragma unroll
    for (int i = 0; i < 6; ++i) r += b[i] * b[i];
    r += gH[0] * gH[0] + gH[1] * gH[1] + gA[0] * gA[0] + gA[1] * gA[1];
#pragma unroll
    for (int i = 0; i < 4; ++i) r += dH[i] * dH[i] + dA[i] * dA[i];
    acc += 0.01f * r;  // LAMBDA_REG
  }
  float s = block_sum256(acc, s_part);   // WMMA wave reduce inside
  if (threadIdx.x == 0) out[0] = s;
}

extern "C" void kernel_launch(void* const* d_in, const int* in_sizes, int n_in,
                              void* d_out, int out_size, void* d_ws, size_t ws_size,
                              hipStream_t stream) {
  const float* a_H      = (const float*)d_in[0];
  const float* a_A      = (const float*)d_in[1];
  const float* a_H_init = (const float*)d_in[2];
  const float* a_A_init = (const float*)d_in[3];
  const float* b        = (const float*)d_in[4];
  const float* gamma_H  = (const float*)d_in[5];
  const float* gamma_A  = (const float*)d_in[6];
  const float* delta_H  = (const float*)d_in[7];
  const float* delta_A  = (const float*)d_in[8];
  const int*   bmi      = (const int*)d_in[9];
  const int*   bbi      = (const int*)d_in[10];
  const int*   bsi      = (const int*)d_in[11];
  const int*   bdi      = (const int*)d_in[12];
  const float* bdur     = (const float*)d_in[13];
  const int*   gmi      = (const int*)d_in[14];
  const int*   gbi      = (const int*)d_in[15];
  const int*   gsi      = (const int*)d_in[16];
  const int*   gdi      = (const int*)d_in[17];
  const unsigned char* gih = (const unsigned char*)d_in[18];

  const int M  = in_sizes[0];
  const int NB = in_sizes[9];
  const int NG = in_sizes[14];

  int blocks = 1024;
  size_t maxb = ws_size / sizeof(float);
  if (maxb < (size_t)blocks) blocks = (int)(maxb > 0 ? maxb : 1);
  float* block_part = (float*)d_ws;

  mmpp_partial<<<blocks, 256, 0, stream>>>(
      a_H, a_A, a_H_init, a_A_init, b, gamma_H, gamma_A, delta_H, delta_A,
      bmi, bbi, bsi, bdi, bdur, gmi, gbi, gsi, gdi, gih,
      NB, NG, M, block_part);

  mmpp_final<<<1, 256, 0, stream>>>(block_part, blocks, b, gamma_H, gamma_A,
                                    delta_H, delta_A, (float*)d_out);
}